// MOELayer_61383672595055
// MI455X (gfx1250) — compile-verified
//
#include <hip/hip_runtime.h>

#define NE 8
#define IN_FEAT 1024
#define OUT_FEAT 4096
#define N_TOKENS 4096

#define M_TILE 64           // tokens per tile (4 WMMA M-frags per wave)
#define N_BLOCK 512         // columns per workgroup (8 waves x 64 cols)
#define K_CHUNK 128         // K staging chunk in LDS
#define K_STRIDE (K_CHUNK + 4)  // pad: bank = (4r + koff) % 64 -> conflict-free
#define MAX_TILES (N_TOKENS / M_TILE + NE)  // 72 worst case (partial tile per expert)

// ws layout (ints)
#define WS_COUNT 0      // [0..7]   per-expert counts
#define WS_CURS 8       // [8..15]  scatter cursors (= exclusive-scan offsets)
#define WS_NTILES 16    // [16]     number of M-tiles
#define WS_TE 24        // [24..)   tile -> expert
#define WS_TS (24 + MAX_TILES)          // tile -> row start in perm
#define WS_TR (24 + 2 * MAX_TILES)      // tile -> valid rows
#define WS_PERM (24 + 3 * MAX_TILES)    // [.. +4096) token permutation

typedef float v2f __attribute__((ext_vector_type(2)));
typedef float v8f __attribute__((ext_vector_type(8)));

__global__ void k_zero(int* ws) {
    if (threadIdx.x < 8) ws[WS_COUNT + threadIdx.x] = 0;
}

__global__ void k_count(const int* __restrict__ gate, int* __restrict__ ws) {
    int n = blockIdx.x * 256 + threadIdx.x;
    if (n < N_TOKENS) atomicAdd(&ws[WS_COUNT + gate[n]], 1);
}

__global__ void k_scan(int* __restrict__ ws) {
    if (threadIdx.x != 0) return;
    int off = 0, ntiles = 0;
    for (int e = 0; e < NE; ++e) {
        int c = ws[WS_COUNT + e];
        ws[WS_CURS + e] = off;
        int t = (c + M_TILE - 1) / M_TILE;
        for (int i = 0; i < t; ++i) {
            ws[WS_TE + ntiles] = e;
            ws[WS_TS + ntiles] = off + i * M_TILE;
            int rem = c - i * M_TILE;
            ws[WS_TR + ntiles] = rem < M_TILE ? rem : M_TILE;
            ++ntiles;
        }
        off += c;
    }
    ws[WS_NTILES] = ntiles;
}

__global__ void k_scatter(const int* __restrict__ gate, int* __restrict__ ws) {
    int n = blockIdx.x * 256 + threadIdx.x;
    if (n < N_TOKENS) {
        int e = gate[n];
        int p = atomicAdd(&ws[WS_CURS + e], 1);
        ws[WS_PERM + p] = n;
    }
}

// One workgroup: 64 gathered token rows x 512 expert-weight columns.
// 8 waves; each wave: 4 M-frags x 4 N-frags of 16x16, K-loop in steps of 4
// with V_WMMA_F32_16X16X4_F32. Each B frag feeds 4 WMMAs.
__launch_bounds__(256)
__global__ void k_moe_gemm(const float* __restrict__ inp,
                           const float* __restrict__ weight,
                           const int* __restrict__ ws,
                           float* __restrict__ out) {
    __shared__ float As[M_TILE][K_STRIDE];   // 64 x 132 f32 ~ 33 KB
    __shared__ int toks[M_TILE];

    const int ntiles = ws[WS_NTILES];
    const int tile = blockIdx.x;
    if (tile >= ntiles) return;   // uniform across block -> EXEC stays full

    const int e      = ws[WS_TE + tile];
    const int rstart = ws[WS_TS + tile];
    const int rows   = ws[WS_TR + tile];
    const int tid = threadIdx.x;

    if (tid < M_TILE) {
        // padded rows point at a valid token; their results are never stored
        int r = tid;
        toks[r] = (r < rows) ? ws[WS_PERM + rstart + r] : ws[WS_PERM + rstart];
    }

    const int lane = tid & 31;
    const int wave = tid >> 5;
    const int hi   = lane >> 4;   // lane half selects K 0-1 vs 2-3 (ISA A/B layout)
    const int l15  = lane & 15;
    const int nbase = blockIdx.y * N_BLOCK + wave * 64;
    const float* wbase = weight + (size_t)e * OUT_FEAT * IN_FEAT;

    v8f acc[4][4];
#pragma unroll
    for (int m = 0; m < 4; ++m)
#pragma unroll
        for (int j = 0; j < 4; ++j) acc[m][j] = (v8f)0.0f;

    for (int kc = 0; kc < IN_FEAT; kc += K_CHUNK) {
        __syncthreads();  // protect As reuse (also orders toks[] on first pass)
        // Stage 64 rows x 128 floats of gathered A: 2048 float4 / 256 threads
#pragma unroll
        for (int i = 0; i < (M_TILE * K_CHUNK / 4) / 256; ++i) {
            int id = tid + 256 * i;
            int r  = id >> 5;        // 32 float4 per row
            int c4 = id & 31;
            float4 v = *(const float4*)(inp + (size_t)toks[r] * IN_FEAT + kc + c4 * 4);
            *(float4*)(&As[r][c4 * 4]) = v;
        }
        __syncthreads();

        for (int k = 0; k < K_CHUNK; k += 4) {
            const int koff = k + hi * 2;
            // A frags: lane half 0 -> K k..k+1, half 1 -> K k+2..k+3 (row = l15 + 16m)
            v2f a[4];
#pragma unroll
            for (int m = 0; m < 4; ++m)
                a[m] = *(const v2f*)(&As[m * 16 + l15][koff]);
#pragma unroll
            for (int j = 0; j < 4; ++j) {
                const int col = nbase + j * 16 + l15;
                // B frag: B[k][n] = W[n][k]; 8B contiguous per lane, reused x4
                v2f b = *(const v2f*)(wbase + (size_t)col * IN_FEAT + (kc + koff));
#pragma unroll
                for (int m = 0; m < 4; ++m)
                    acc[m][j] = __builtin_amdgcn_wmma_f32_16x16x4_f32(
                        false, a[m], false, b, (short)0, acc[m][j], false, false);
            }
        }
    }

    // C/D layout: VGPR g holds row g (lanes 0-15) / row g+8 (lanes 16-31), col = lane%16
#pragma unroll
    for (int m = 0; m < 4; ++m)
#pragma unroll
        for (int j = 0; j < 4; ++j) {
            const int col = nbase + j * 16 + l15;
#pragma unroll
            for (int g = 0; g < 8; ++g) {
                const int r = m * 16 + g + hi * 8;
                if (r < rows)
                    out[(size_t)toks[r] * OUT_FEAT + col] = acc[m][j][g];
            }
        }
}

extern "C" void kernel_launch(void* const* d_in, const int* in_sizes, int n_in,
                              void* d_out, int out_size, void* d_ws, size_t ws_size,
                              hipStream_t stream) {
    const float* inp    = (const float*)d_in[0];
    const int*   gate   = (const int*)d_in[1];
    const float* weight = (const float*)d_in[2];
    float* out = (float*)d_out;
    int*   ws  = (int*)d_ws;

    k_zero<<<1, 32, 0, stream>>>(ws);
    k_count<<<(N_TOKENS + 255) / 256, 256, 0, stream>>>(gate, ws);
    k_scan<<<1, 32, 0, stream>>>(ws);
    k_scatter<<<(N_TOKENS + 255) / 256, 256, 0, stream>>>(gate, ws);

    dim3 grid(MAX_TILES, OUT_FEAT / N_BLOCK);
    k_moe_gemm<<<grid, 256, 0, stream>>>(inp, weight, ws, out);
}